// LDGCNN_9801115369860
// MI455X (gfx1250) — compile-verified
//
#include <hip/hip_runtime.h>
#include <hip/hip_bf16.h>

// ---------------- problem constants ----------------
constexpr int NB = 16;      // batch
constexpr int NP = 2048;    // points
constexpr int NK = 20;      // neighbors
constexpr int D0 = 64;      // edge conv out
constexpr int D1 = 64;
constexpr int D2 = 128;
constexpr int D3 = 256;
constexpr int DCAT = 512;   // 64+64+128+256
constexpr int DF = 1024;    // final dim
constexpr int TM = 64;      // rows per block in finalK (weight reuse factor 4)
constexpr float EPS = 1e-5f;
constexpr float SLOPE = 0.2f;

typedef __attribute__((ext_vector_type(2))) float v2f;
typedef __attribute__((ext_vector_type(8))) float v8f;

// order-preserving float <-> uint32 encoding (for atomicMax-based fp max)
__device__ __forceinline__ unsigned encf(float f) {
    unsigned u = __float_as_uint(f);
    return (u & 0x80000000u) ? ~u : (u | 0x80000000u);
}
__device__ __forceinline__ float decf(unsigned e) {
    unsigned u = (e & 0x80000000u) ? (e & 0x7FFFFFFFu) : ~e;
    return __uint_as_float(u);
}
__device__ __forceinline__ float lrelu(float x) { return x > 0.f ? x : SLOPE * x; }

// ---------------- init: zero BN accumulators, set -inf encodings ----------------
__global__ void initK(float* sums, unsigned* enc) {
    int i = blockIdx.x * blockDim.x + threadIdx.x;
    if (i < 1024) sums[i] = 0.f;               // gsum[512] + gsumsq[512]
    if (i < NB * DF) enc[i] = 0x007FFFFFu;     // encf(-inf)
}

// ---------------- kNN: brute force, points in LDS, top-20 insertion sort ----------------
__global__ __launch_bounds__(64) void knnK(const float* __restrict__ x, int* __restrict__ idx) {
    __shared__ float px[NP], py[NP], pz[NP];   // 24 KB
    const int tid = threadIdx.x, bid = blockIdx.x;
    const int b = bid / (NP / 64);
    const int n = (bid % (NP / 64)) * 64 + tid;
    const float* xb = x + (size_t)b * NP * 3;
    for (int m = tid; m < NP; m += 64) {
        px[m] = xb[m * 3 + 0]; py[m] = xb[m * 3 + 1]; pz[m] = xb[m * 3 + 2];
    }
    __syncthreads();
    float dk[NK]; int ik[NK];
#pragma unroll
    for (int k = 0; k < NK; ++k) { dk[k] = 3.4e38f; ik[k] = 0; }
    const float qx = px[n], qy = py[n], qz = pz[n];
    for (int m = 0; m < NP; ++m) {
        float dx = px[m] - qx, dy = py[m] - qy, dz = pz[m] - qz;
        float d = dx * dx + dy * dy + dz * dz;
        if (d < dk[NK - 1]) {                  // swap-down insertion, fully unrolled
            float cd = d; int ci = m;
#pragma unroll
            for (int k = 0; k < NK; ++k) {
                if (cd < dk[k]) {
                    float td = dk[k]; int ti = ik[k];
                    dk[k] = cd; ik[k] = ci; cd = td; ci = ti;
                }
            }
        }
    }
    int* op = idx + ((size_t)b * NP + n) * NK;
#pragma unroll
    for (int k = 0; k < NK; ++k) op[k] = ik[k];
}

// ---------------- edge conv: h0_pre = max_k(W_e @ [nb-c, c]); fused BN stats ----------------
__global__ __launch_bounds__(64) void edgeK(const float* __restrict__ x, const int* __restrict__ idx,
                                            const float* __restrict__ wedge, float* __restrict__ h0,
                                            float* __restrict__ gsum, float* __restrict__ gsumsq) {
    __shared__ float sw[D0 * 6];
    __shared__ float lsum[D0], lsq[D0];
    const int tid = threadIdx.x, bid = blockIdx.x;
    const int b = bid / (NP / 64);
    const int n = (bid % (NP / 64)) * 64 + tid;
    for (int e = tid; e < D0 * 6; e += 64) sw[e] = wedge[e];
    lsum[tid] = 0.f; lsq[tid] = 0.f;
    __syncthreads();
    const float* xb = x + (size_t)b * NP * 3;
    const float cx = xb[n * 3 + 0], cy = xb[n * 3 + 1], cz = xb[n * 3 + 2];
    float nbx[NK], nby[NK], nbz[NK];
#pragma unroll
    for (int k = 0; k < NK; ++k) {
        int j = idx[((size_t)b * NP + n) * NK + k];
        nbx[k] = xb[j * 3 + 0] - cx; nby[k] = xb[j * 3 + 1] - cy; nbz[k] = xb[j * 3 + 2] - cz;
    }
    for (int c = 0; c < D0; ++c) {
        const float w0 = sw[c * 6 + 0], w1 = sw[c * 6 + 1], w2 = sw[c * 6 + 2];
        const float base = sw[c * 6 + 3] * cx + sw[c * 6 + 4] * cy + sw[c * 6 + 5] * cz;
        float m = -3.4e38f, s = 0.f, ss = 0.f;
#pragma unroll
        for (int k = 0; k < NK; ++k) {
            float a = w0 * nbx[k] + w1 * nby[k] + w2 * nbz[k] + base;
            m = fmaxf(m, a); s += a; ss += a * a;
        }
        h0[((size_t)b * NP + n) * D0 + c] = m;
        atomicAdd(&lsum[c], s); atomicAdd(&lsq[c], ss);
    }
    __syncthreads();
    atomicAdd(&gsum[tid], lsum[tid]);
    atomicAdd(&gsumsq[tid], lsq[tid]);
}

// ---------------- per-channel BN scale/shift from accumulated sums ----------------
__global__ void statsK(const float* __restrict__ gsum, const float* __restrict__ gsumsq,
                       float* __restrict__ scale, float* __restrict__ shift,
                       int base, int nchan, float count) {
    int c = blockIdx.x * blockDim.x + threadIdx.x;
    if (c >= nchan) return;
    int gc = base + c;
    float mean = gsum[gc] / count;
    float var = gsumsq[gc] / count - mean * mean;
    float sc = rsqrtf(var + EPS);
    scale[gc] = sc;
    shift[gc] = -mean * sc;
}

// ---------------- graph layer: gather-max -> BN/LReLU -> f32 WMMA GEMM + fused stats ----------------
template <int CIN, int COUT>
__global__ __launch_bounds__(128) void layerK(const float* __restrict__ hin, float* __restrict__ hout,
                                              const int* __restrict__ idx, const float* __restrict__ w,
                                              const float* __restrict__ scale, const float* __restrict__ shift,
                                              float* __restrict__ gsum, float* __restrict__ gsumsq) {
    __shared__ int sidx[16 * NK];
    __shared__ alignas(16) float sa[16][CIN];
    __shared__ float lsum[COUT], lsq[COUT];
    const int tid = threadIdx.x, bid = blockIdx.x;
    const int b = bid / (NP / 16);
    const int n0 = (bid % (NP / 16)) * 16;

    for (int c = tid; c < COUT; c += 128) { lsum[c] = 0.f; lsq[c] = 0.f; }
    for (int e = tid; e < 16 * NK; e += 128) {
        int r = e / NK, k = e % NK;
        sidx[e] = idx[((size_t)b * NP + n0 + r) * NK + k];
    }
    __syncthreads();

    // gather neighbors of previous layer's pre-BN activations, max, then normalize
    for (int e = tid; e < 16 * CIN; e += 128) {
        int r = e / CIN, c = e % CIN;
        const float* hb = hin + (size_t)b * NP * CIN + c;
        float g = -3.4e38f;
#pragma unroll
        for (int k = 0; k < NK; ++k) {
            int j = sidx[r * NK + k];
            g = fmaxf(g, hb[(size_t)j * CIN]);
        }
        sa[r][c] = lrelu(scale[c] * g + shift[c]);
    }
    __syncthreads();

    const int lane = tid & 31, wv = tid >> 5;
    const int l15 = lane & 15, hi = lane >> 4;
    for (int ot = wv; ot < COUT / 16; ot += 4) {
        const int obase = ot * 16;
        v8f acc = {};
#pragma unroll 4
        for (int k0 = 0; k0 < CIN; k0 += 4) {
            v2f av = *reinterpret_cast<const v2f*>(&sa[l15][k0 + 2 * hi]);
            v2f bv = *reinterpret_cast<const v2f*>(&w[(size_t)(obase + l15) * CIN + k0 + 2 * hi]);
            acc = __builtin_amdgcn_wmma_f32_16x16x4_f32(false, av, false, bv, (short)0, acc, false, false);
        }
        const int o = obase + l15;
        float s = 0.f, ss = 0.f;
#pragma unroll
        for (int r = 0; r < 8; ++r) {
            int m = r + 8 * hi;
            float v = acc[r];
            hout[((size_t)b * NP + n0 + m) * COUT + o] = v;
            s += v; ss += v * v;
        }
        atomicAdd(&lsum[o], s); atomicAdd(&lsq[o], ss);
    }
    __syncthreads();
    for (int c = tid; c < COUT; c += 128) {
        atomicAdd(&gsum[c], lsum[c]);
        atomicAdd(&gsumsq[c], lsq[c]);
    }
}

// ---------------- final projection: WMMA GEMM (N_pts x 1024 x 512) + fused max over N ----------------
// 64 rows per block: each weight fetch feeds 4 independent WMMA accumulators
// (w_final L2 traffic: 1 GB instead of 4 GB -> compute-bound, not L2-bound).
__global__ __launch_bounds__(256) void finalK(const float* __restrict__ h0, const float* __restrict__ h1,
                                              const float* __restrict__ h2, const float* __restrict__ h3,
                                              const float* __restrict__ scale, const float* __restrict__ shift,
                                              const float* __restrict__ wf, unsigned* __restrict__ outEnc) {
    __shared__ alignas(16) float sa[TM][DCAT];  // 64 x 512 x 4B = 128 KB (of 320 KB/WGP)
    const int tid = threadIdx.x, bid = blockIdx.x;
    const int b = bid / (NP / TM);
    const int n0 = (bid % (NP / TM)) * TM;

    for (int e = tid; e < TM * DCAT; e += 256) {
        int r = e >> 9, c = e & 511;
        float pre;
        size_t row = (size_t)b * NP + n0 + r;
        if (c < 64)       pre = h0[row * D0 + c];
        else if (c < 128) pre = h1[row * D1 + (c - 64)];
        else if (c < 256) pre = h2[row * D2 + (c - 128)];
        else              pre = h3[row * D3 + (c - 256)];
        sa[r][c] = lrelu(scale[c] * pre + shift[c]);
    }
    __syncthreads();

    const int lane = tid & 31, wv = tid >> 5;
    const int l15 = lane & 15, hi = lane >> 4;
    for (int ot = wv; ot < DF / 16; ot += 8) {
        const int o = ot * 16 + l15;
        const float* wrow = wf + (size_t)o * DCAT;
        v8f acc0 = {}, acc1 = {}, acc2 = {}, acc3 = {};
#pragma unroll 4
        for (int k0 = 0; k0 < DCAT; k0 += 4) {
            const int kk = k0 + 2 * hi;
            v2f bv = *reinterpret_cast<const v2f*>(&wrow[kk]);
            v2f a0 = *reinterpret_cast<const v2f*>(&sa[l15][kk]);
            v2f a1 = *reinterpret_cast<const v2f*>(&sa[16 + l15][kk]);
            v2f a2 = *reinterpret_cast<const v2f*>(&sa[32 + l15][kk]);
            v2f a3 = *reinterpret_cast<const v2f*>(&sa[48 + l15][kk]);
            acc0 = __builtin_amdgcn_wmma_f32_16x16x4_f32(false, a0, false, bv, (short)0, acc0, false, false);
            acc1 = __builtin_amdgcn_wmma_f32_16x16x4_f32(false, a1, false, bv, (short)0, acc1, false, false);
            acc2 = __builtin_amdgcn_wmma_f32_16x16x4_f32(false, a2, false, bv, (short)0, acc2, false, false);
            acc3 = __builtin_amdgcn_wmma_f32_16x16x4_f32(false, a3, false, bv, (short)0, acc3, false, false);
        }
        float m = fmaxf(fmaxf(acc0[0], acc1[0]), fmaxf(acc2[0], acc3[0]));
#pragma unroll
        for (int r = 1; r < 8; ++r) {
            m = fmaxf(m, fmaxf(fmaxf(acc0[r], acc1[r]), fmaxf(acc2[r], acc3[r])));
        }
        atomicMax(&outEnc[b * DF + o], encf(m));   // both half-lanes hit the same o: merged by atomic
    }
}

// ---------------- decode max + add final bias ----------------
__global__ void decodeK(const unsigned* __restrict__ enc, const float* __restrict__ bf,
                        float* __restrict__ out) {
    int i = blockIdx.x * blockDim.x + threadIdx.x;
    if (i < NB * DF) out[i] = decf(enc[i]) + bf[i & (DF - 1)];
}

extern "C" void kernel_launch(void* const* d_in, const int* in_sizes, int n_in,
                              void* d_out, int out_size, void* d_ws, size_t ws_size,
                              hipStream_t stream) {
    (void)in_sizes; (void)n_in; (void)out_size; (void)ws_size;
    const float* x      = (const float*)d_in[0];
    const float* w_edge = (const float*)d_in[1];
    const float* w1     = (const float*)d_in[3];
    const float* w2     = (const float*)d_in[5];
    const float* w3     = (const float*)d_in[7];
    const float* wf     = (const float*)d_in[9];
    const float* bf     = (const float*)d_in[10];
    float* out = (float*)d_out;

    // workspace layout
    char* ws = (char*)d_ws;
    size_t off = 0;
    int*   idx  = (int*)(ws + off);   off += (size_t)NB * NP * NK * 4;          // 2.5 MB
    float* h0   = (float*)(ws + off); off += (size_t)NB * NP * D0 * 4;          // 8 MB
    float* h1   = (float*)(ws + off); off += (size_t)NB * NP * D1 * 4;          // 8 MB
    float* h2   = (float*)(ws + off); off += (size_t)NB * NP * D2 * 4;          // 16 MB
    float* h3   = (float*)(ws + off); off += (size_t)NB * NP * D3 * 4;          // 32 MB
    float* gsum   = (float*)(ws + off); off += 512 * 4;
    float* gsumsq = (float*)(ws + off); off += 512 * 4;
    float* scale  = (float*)(ws + off); off += 512 * 4;
    float* shift  = (float*)(ws + off); off += 512 * 4;
    unsigned* outEnc = (unsigned*)(ws + off); off += (size_t)NB * DF * 4;

    const float CNT0 = (float)NB * NP * NK;   // edge conv BN count (B,N,k)
    const float CNT1 = (float)NB * NP;        // layer BN count (B,N)

    initK<<<64, 256, 0, stream>>>(gsum, outEnc);
    knnK<<<NB * (NP / 64), 64, 0, stream>>>(x, idx);
    edgeK<<<NB * (NP / 64), 64, 0, stream>>>(x, idx, w_edge, h0, gsum + 0, gsumsq + 0);
    statsK<<<1, 64, 0, stream>>>(gsum, gsumsq, scale, shift, 0, D0, CNT0);

    layerK<64, 64><<<NB * NP / 16, 128, 0, stream>>>(h0, h1, idx, w1, scale + 0, shift + 0,
                                                     gsum + 64, gsumsq + 64);
    statsK<<<1, 64, 0, stream>>>(gsum, gsumsq, scale, shift, 64, D1, CNT1);

    layerK<64, 128><<<NB * NP / 16, 128, 0, stream>>>(h1, h2, idx, w2, scale + 64, shift + 64,
                                                      gsum + 128, gsumsq + 128);
    statsK<<<1, 128, 0, stream>>>(gsum, gsumsq, scale, shift, 128, D2, CNT1);

    layerK<128, 256><<<NB * NP / 16, 128, 0, stream>>>(h2, h3, idx, w3, scale + 128, shift + 128,
                                                       gsum + 256, gsumsq + 256);
    statsK<<<1, 256, 0, stream>>>(gsum, gsumsq, scale, shift, 256, D3, CNT1);

    finalK<<<NB * NP / TM, 256, 0, stream>>>(h0, h1, h2, h3, scale, shift, wf, outEnc);
    decodeK<<<(NB * DF + 255) / 256, 256, 0, stream>>>(outEnc, bf, out);
}